// ContrastiveLoss_56435870269983
// MI455X (gfx1250) — compile-verified
//
#include <hip/hip_runtime.h>

typedef __attribute__((ext_vector_type(2))) float v2f;
typedef __attribute__((ext_vector_type(8))) float v8f;
typedef __attribute__((ext_vector_type(4))) int   v4i;
typedef __attribute__((address_space(1))) v4i*    as1_v4i;
typedef __attribute__((address_space(3))) v4i*    as3_v4i;

#define CL_D        64          // embedding dim (K of the GEMM)
#define CL_LDSROW   68          // padded LDS row stride in floats (bank-conflict free)
#define CL_TTRACKS  512         // number of tracks (power of 2 -> mask with &)
#define CL_NROWT    512         // 8192 x-rows / 16
#define CL_NCOLS    4096        // T*Q y rows
#define CL_COLSPT   128         // columns per LDS supertile (8 waves x 16 cols)
#define CL_NST      (CL_NCOLS / CL_COLSPT)   // 32 supertiles

// CDNA5 async global->LDS path (no VGPR staging, tracked by ASYNCcnt).
#if defined(__AMDGCN__) && __has_builtin(__builtin_amdgcn_global_load_async_to_lds_b128) && \
    __has_builtin(__builtin_amdgcn_s_wait_asynccnt)
#define CL_ASYNC 1
#else
#define CL_ASYNC 0
#endif

#if CL_ASYNC
// Each thread moves 8 float4 chunks of one 128x64 Y supertile straight into
// (padded) LDS via global_load_async_to_lds_b128 (tracked by ASYNCcnt).
__device__ __forceinline__ void cl_issue_async(const float* __restrict__ ysrc,
                                               float* ldsbuf, int tid) {
#pragma unroll
  for (int i = 0; i < 8; ++i) {
    const int q   = tid + 256 * i;       // float4 chunk index in supertile
    const int row = q >> 4;              // 16 chunks per 64-float row
    const int c   = (q & 15) << 2;
    __builtin_amdgcn_global_load_async_to_lds_b128(
        (as1_v4i)(const void*)(ysrc + 4 * q),
        (as3_v4i)(void*)&ldsbuf[row * CL_LDSROW + c],
        0, 0);
  }
}
#endif

// ---------------------------------------------------------------------------
// Main kernel: one block (256 thr = 8 waves) owns a 16-row x tile and sweeps
// all 4096 y columns. Each wave computes 16x16 f32 tiles with
// v_wmma_f32_16x16x4_f32 (16 chained WMMAs over K=64), applies exp()/mask in
// the epilogue and accumulates num/den. Deterministic: per-block partials.
// ---------------------------------------------------------------------------
__global__ __launch_bounds__(256) void cl_gemm_kernel(
    const float* __restrict__ x, const int* __restrict__ tidx,
    const float* __restrict__ y, float* __restrict__ partials)
{
  __shared__ float lx[16 * CL_LDSROW];                 // x tile, padded
#if CL_ASYNC
  __shared__ float ly[2][CL_COLSPT * CL_LDSROW];       // double-buffered y supertile
#else
  __shared__ float ly[1][CL_COLSPT * CL_LDSROW];
#endif
  __shared__ float red[16];                            // 8 waves x {num,den}

  const int tid  = threadIdx.x;
  const int wave = tid >> 5;
  const int lane = tid & 31;
  const int h    = lane >> 4;     // lane half (K split for A/B fragments)
  const int li   = lane & 15;     // index within half (M for A, N for B)
  const int r0   = blockIdx.x << 4;

  // ---- stage X row-tile (16 x 64 f32) into padded LDS, coalesced float4 ----
  {
    const int row = tid >> 4;             // 16 float4 chunks per row
    const int c   = (tid & 15) << 2;
    float4 v = ((const float4*)(x + r0 * CL_D))[tid];
    *(float4*)&lx[row * CL_LDSROW + c] = v;
  }

  // track ids of this lane's 8 C-rows (C layout: M = v + 8*h, N = li)
  int tv[8];
#pragma unroll
  for (int v = 0; v < 8; ++v) tv[v] = tidx[r0 + v + 8 * h];

#if CL_ASYNC
  cl_issue_async(y, ly[0], tid);          // prime the pipeline with supertile 0
#endif

  __syncthreads();

  // ---- A fragments for all 16 K-steps: lane holds x[li][4s+2h .. +1] ----
  v2f afrag[16];
#pragma unroll
  for (int s = 0; s < 16; ++s)
    afrag[s] = *(const v2f*)&lx[li * CL_LDSROW + 4 * s + 2 * h];

  float num = 0.0f, den = 0.0f;
  const float invT = 1.0f / 0.3f;

  for (int st = 0; st < CL_NST; ++st) {
#if CL_ASYNC
    const int cur = st & 1;
    if (st + 1 < CL_NST) {
      // issue next supertile into the other buffer; overlaps the WMMA chain
      cl_issue_async(y + (st + 1) * CL_COLSPT * CL_D, ly[cur ^ 1], tid);
      __builtin_amdgcn_s_wait_asynccnt(8);   // my batch-st loads have landed
    } else {
      __builtin_amdgcn_s_wait_asynccnt(0);
    }
    __syncthreads();                         // everyone's batch-st data visible
    const float* by = &ly[cur][(wave * 16 + li) * CL_LDSROW + 2 * h];
#else
    // fallback: transient global->VGPR->LDS copy (no long-lived stage array)
    {
      const float4* src = (const float4*)(y + st * CL_COLSPT * CL_D);
#pragma unroll
      for (int i = 0; i < 8; ++i) {
        const int q   = tid + 256 * i;
        const int row = q >> 4;
        const int c   = (q & 15) << 2;
        *(float4*)&ly[0][row * CL_LDSROW + c] = src[q];
      }
    }
    __syncthreads();
    const float* by = &ly[0][(wave * 16 + li) * CL_LDSROW + 2 * h];
#endif

    const int c0 = st * CL_COLSPT + wave * 16;

    // D = A x B + C over K=64 via 16 chained v_wmma_f32_16x16x4_f32
    v8f acc = {};
#pragma unroll
    for (int s = 0; s < 16; ++s) {
      v2f b = *(const v2f*)(by + 4 * s);
      acc = __builtin_amdgcn_wmma_f32_16x16x4_f32(
          /*neg_a=*/false, afrag[s], /*neg_b=*/false, b,
          /*c_mod=*/(short)0, acc, /*reuse_a=*/false, /*reuse_b=*/false);
    }

    // epilogue: exp-scale, mask on track id, accumulate num/den
#pragma unroll
    for (int v = 0; v < 8; ++v) {
      float sv = __expf(acc[v] * invT);
      int j = (tv[v] - c0) & (CL_TTRACKS - 1);   // matches iff (c0+li)%512 == t
      if (j == li) num += sv; else den += sv;
    }

    __syncthreads();   // all waves done reading this buffer before it is refilled
  }

  // wave32 reduction
#pragma unroll
  for (int off = 16; off > 0; off >>= 1) {
    num += __shfl_xor(num, off, 32);
    den += __shfl_xor(den, off, 32);
  }
  if (lane == 0) { red[wave * 2] = num; red[wave * 2 + 1] = den; }
  __syncthreads();
  if (tid == 0) {
    float n = 0.0f, d = 0.0f;
#pragma unroll
    for (int w = 0; w < 8; ++w) { n += red[2 * w]; d += red[2 * w + 1]; }
    partials[blockIdx.x * 2 + 0] = n;
    partials[blockIdx.x * 2 + 1] = d;
  }
}

// ---------------------------------------------------------------------------
// Finalize: deterministic reduction of 512 partial pairs -> scalar loss.
// ---------------------------------------------------------------------------
__global__ __launch_bounds__(256) void cl_finalize_kernel(
    const float* __restrict__ partials, float* __restrict__ out)
{
  __shared__ float sn[256];
  __shared__ float sd[256];
  const int t = threadIdx.x;
  float n = 0.0f, d = 0.0f;
  for (int i = t; i < CL_NROWT; i += 256) {
    n += partials[2 * i + 0];
    d += partials[2 * i + 1];
  }
  sn[t] = n; sd[t] = d;
  __syncthreads();
  for (int s = 128; s > 0; s >>= 1) {
    if (t < s) { sn[t] += sn[t + s]; sd[t] += sd[t + s]; }
    __syncthreads();
  }
  if (t == 0) {
    float num = sn[0], den = sd[0];
    out[0] = -logf(num / (den + num + 1e-9f) + 1e-10f);
  }
}

extern "C" void kernel_launch(void* const* d_in, const int* in_sizes, int n_in,
                              void* d_out, int out_size, void* d_ws, size_t ws_size,
                              hipStream_t stream) {
  const float* x    = (const float*)d_in[0];   // (8192, 64) f32
  const int*   tidx = (const int*)d_in[1];     // (8192,) int
  const float* y    = (const float*)d_in[2];   // (512, 8, 64) f32 -> flat 4096x64
  float* partials   = (float*)d_ws;            // 512 * 2 floats
  (void)in_sizes; (void)n_in; (void)out_size; (void)ws_size;

  cl_gemm_kernel<<<CL_NROWT, 256, 0, stream>>>(x, tidx, y, partials);
  cl_finalize_kernel<<<1, 256, 0, stream>>>(partials, (float*)d_out);
}